// uBRU_28664611733940
// MI455X (gfx1250) — compile-verified
//
#include <hip/hip_runtime.h>
#include <math.h>

typedef __attribute__((ext_vector_type(2))) float v2f;
typedef __attribute__((ext_vector_type(4))) float v4f;
typedef __attribute__((ext_vector_type(8))) float v8f;

#define LN_EPS   1e-5f
#define REC_EPS  1e-11f
#define CLAMPMAX 10.0f

static constexpr int Bb = 32, Tt = 2048, Dd = 256, Hh = 256;
static constexpr int MROWS = Bb * Tt;      // 65536 rows
static constexpr int NCH   = Bb * Hh;      // 8192 scan channels
static constexpr int CC    = 64;           // chunks per sequence
static constexpr int LL    = Tt / CC;      // 32 steps per chunk

__device__ __forceinline__ float frcp(float x) {
#if __has_builtin(__builtin_amdgcn_rcpf)
    return __builtin_amdgcn_rcpf(x);       // v_rcp_f32
#else
    return 1.0f / x;
#endif
}
__device__ __forceinline__ float sigm(float x) {
    return frcp(1.0f + __expf(-x));
}

// ---------------------------------------------------------------------------
// Fused GEMM (X @ W^T) + LayerNorm(g,b) + r = exp(min(.,10))
// Each block: 8 waves, 128 rows; each wave: 16 rows x 256 cols via
// 16 v_wmma_f32_16x16x4_f32 accumulator tiles. W staged in LDS (256 KB)
// with async global->LDS b128 copies tracked by ASYNCcnt.
// ---------------------------------------------------------------------------
__global__ __launch_bounds__(256, 1)
void gemm_ln_exp_kernel(const float* __restrict__ X,
                        const float* __restrict__ W,
                        const float* __restrict__ g,
                        const float* __restrict__ bias,
                        float* __restrict__ rout)
{
    extern __shared__ float ldsW[];           // Hh * Dd floats = 256 KB
    const int tid  = threadIdx.x;
    const int lane = tid & 31;
    const int wave = tid >> 5;
    const int l15  = lane & 15;
    const int half = lane >> 4;

    // Async stage of W into LDS. Generic LDS pointer low 32 bits = LDS byte
    // address (flat aperture keeps the offset in addr[31:0]).
    {
        const unsigned ldsb = (unsigned)(size_t)(void*)ldsW;
        const unsigned long long gb = (unsigned long long)(size_t)W;
        const int nvec = (Hh * Dd) / 4;       // 16384 x b128
        for (int i = tid; i < nvec; i += 256) {
            unsigned la = ldsb + (unsigned)i * 16u;
            unsigned long long ga = gb + (unsigned long long)i * 16ull;
            asm volatile("global_load_async_to_lds_b128 %0, %1, off"
                         :: "v"(la), "v"(ga) : "memory");
        }
        asm volatile("s_wait_asynccnt 0" ::: "memory");
    }
    __syncthreads();

    const int m0 = blockIdx.x * 128 + wave * 16;

    v8f acc[16];
    v8f zero = {};
#pragma unroll
    for (int n = 0; n < 16; ++n) acc[n] = zero;

    // A fragment (32-bit 16x4): lane -> M = lane&15, K = {0,1}/{2,3} per half.
    const float* arow = X + (size_t)(m0 + l15) * Dd + 2 * half;

    for (int k0 = 0; k0 < Dd; k0 += 4) {
        v2f afrag = *(const v2f*)(arow + k0);
#pragma unroll
        for (int n = 0; n < 16; ++n) {
            const float* bptr = ldsW + (n * 16 + l15) * Dd + k0 + 2 * half;
            v2f bfrag = *(const v2f*)bptr;                  // ds_load_b64
            acc[n] = __builtin_amdgcn_wmma_f32_16x16x4_f32(
                false, afrag, false, bfrag, (short)0, acc[n], false, false);
        }
    }

    // LayerNorm epilogue: row stats = 16-tile register sum + xor butterfly
    // within each 16-lane half (rows v / v+8).
    float mu[8], rstd[8];
#pragma unroll
    for (int v = 0; v < 8; ++v) {
        float s = 0.f, ss = 0.f;
#pragma unroll
        for (int n = 0; n < 16; ++n) {
            float xv = acc[n][v];
            s += xv; ss += xv * xv;
        }
#pragma unroll
        for (int m = 1; m <= 8; m <<= 1) {
            s  += __shfl_xor(s,  m, 32);
            ss += __shfl_xor(ss, m, 32);
        }
        float mean = s * (1.0f / Hh);
        float var  = ss * (1.0f / Hh) - mean * mean;
        mu[v]   = mean;
        rstd[v] = rsqrtf(var + LN_EPS);
    }

    const int rowb = m0 + half * 8;
#pragma unroll
    for (int n = 0; n < 16; ++n) {
        const int col = n * 16 + l15;
        const float gg = g[col];
        const float bb = bias[col];
#pragma unroll
        for (int v = 0; v < 8; ++v) {
            float val = (acc[n][v] - mu[v]) * rstd[v] * gg + bb;
            val = fminf(val, CLAMPMAX);
            rout[(size_t)(rowb + v) * Hh + col] = __expf(val);
        }
    }
}

// ---------------------------------------------------------------------------
// Chunked parallel scans. Forward step a' = p/(p + r(1-p)+eps), p = ca*a+cb
// is a Moebius map -> 2x2 matrix composition (renormalized each step).
// Backward step h' = u*h + v is affine -> projective (U,V,W) composition.
// Phase 1 kernels run one thread per (chunk, channel): 524288 lanes.
// ---------------------------------------------------------------------------

// F1: per-(chunk,channel) forward Moebius composition over the chunk.
__global__ __launch_bounds__(256)
void fwd_compose_kernel(const float* __restrict__ r,
                        const float* __restrict__ probs,
                        float* __restrict__ M)     // [4][CC][NCH]
{
    const int tid = blockIdx.x * 256 + threadIdx.x;
    const int ch  = tid & (NCH - 1);
    const int c   = tid >> 13;                     // NCH = 2^13
    const int h   = ch & (Hh - 1);
    const int b   = ch >> 8;
    const size_t base = (size_t)b * Tt * Hh + h;
    const float p1 = sigm(probs[Hh + h]);
    const float p2 = sigm(probs[2 * Hh + h]);
    const float ca = p1 - p2, cb = p2;

    float m00 = 1.f, m01 = 0.f, m10 = 0.f, m11 = 1.f;
    const int t0 = c * LL;
    for (int i = 0; i < LL; ++i) {
        const size_t idx = base + (size_t)(t0 + i) * Hh;
        if (i + 8 < LL)
            __builtin_prefetch(&r[idx + (size_t)8 * Hh], 0, 0);
        float rt  = r[idx];
        float gam = ca * (1.0f - rt);
        float del = cb + rt * (1.0f - cb) + REC_EPS;
        float s   = frcp(del);
        float cas = ca * s, cbs = cb * s, gs = gam * s;
        float n00 = cas * m00 + cbs * m10;
        float n01 = cas * m01 + cbs * m11;
        float n10 = gs * m00 + m10;
        float n11 = gs * m01 + m11;
        float s2  = frcp(fabsf(n10) + fabsf(n11) + 1e-30f);
        m00 = n00 * s2; m01 = n01 * s2; m10 = n10 * s2; m11 = n11 * s2;
    }
    M[0 * CC * NCH + c * NCH + ch] = m00;
    M[1 * CC * NCH + c * NCH + ch] = m01;
    M[2 * CC * NCH + c * NCH + ch] = m10;
    M[3 * CC * NCH + c * NCH + ch] = m11;
}

// F2: per-channel serial scan over CC chunk transforms -> chunk entry states.
__global__ __launch_bounds__(256)
void fwd_boundary_kernel(const float* __restrict__ probs,
                         const float* __restrict__ M,
                         float* __restrict__ astart)   // [CC][NCH]
{
    const int ch = blockIdx.x * 256 + threadIdx.x;
    const int h  = ch & (Hh - 1);
    float a = sigm(probs[h]);
    for (int c = 0; c < CC; ++c) {
        astart[c * NCH + ch] = a;
        float m00 = M[0 * CC * NCH + c * NCH + ch];
        float m01 = M[1 * CC * NCH + c * NCH + ch];
        float m10 = M[2 * CC * NCH + c * NCH + ch];
        float m11 = M[3 * CC * NCH + c * NCH + ch];
        a = (m00 * a + m01) * frcp(m10 * a + m11);
    }
}

// F3+B1 fused: re-run the exact forward recurrence within the chunk from its
// entry state (writing p,a), keep chunk-local p,a in registers and compose
// the backward affine transform (U,V,W) for the chunk in the same pass.
__global__ __launch_bounds__(256)
void fwd_apply_bwd_compose_kernel(const float* __restrict__ r,
                                  const float* __restrict__ probs,
                                  const float* __restrict__ astart,
                                  float* __restrict__ pbuf,
                                  float* __restrict__ abuf,
                                  float* __restrict__ N)   // [3][CC][NCH]
{
    const int tid = blockIdx.x * 256 + threadIdx.x;
    const int ch  = tid & (NCH - 1);
    const int c   = tid >> 13;
    const int h   = ch & (Hh - 1);
    const int b   = ch >> 8;
    const size_t base = (size_t)b * Tt * Hh + h;
    const float p1 = sigm(probs[Hh + h]);
    const float p2 = sigm(probs[2 * Hh + h]);
    const float q1 = 1.0f - p1;
    const int t0 = c * LL;

    float pa[LL], aa[LL];
    float at = astart[c * NCH + ch];
#pragma unroll
    for (int i = 0; i < LL; ++i) {
        const size_t idx = base + (size_t)(t0 + i) * Hh;
        float rt = r[idx];
        float pt = p1 * at + p2 * (1.0f - at);
        at = pt * frcp(pt + rt * (1.0f - pt) + REC_EPS);
        pa[i] = pt; aa[i] = at;
        pbuf[idx] = pt;
        abuf[idx] = at;
    }

    // Backward affine compose (descending t). Anchor for the last chunk is
    // a[T-1] itself, so it composes one fewer map.
    float U = 1.f, V = 0.f, Wd = 1.f;
    const int itop = (c == CC - 1) ? (LL - 2) : (LL - 1);
#pragma unroll
    for (int i = LL - 1; i >= 0; --i) {
        if (i > itop) continue;
        float a_t = aa[i];
        // p_{t+1}: local, or computed from local last a at the chunk edge.
        float pn = (i == LL - 1) ? (p1 * aa[LL - 1] + p2 * (1.0f - aa[LL - 1]))
                                 : pa[i + 1];
        float ipn = frcp(pn);
        float iqn = frcp(1.0f - pn);
        float u = a_t * (p1 * ipn - q1 * iqn);
        float v = a_t * (q1 * iqn);
        float nU = u * U;
        float nV = u * V + v * Wd;
        float s  = frcp(fabsf(nU) + fabsf(nV) + Wd + 1e-30f);
        U = nU * s; V = nV * s; Wd = Wd * s;
    }
    N[0 * CC * NCH + c * NCH + ch] = U;
    N[1 * CC * NCH + c * NCH + ch] = V;
    N[2 * CC * NCH + c * NCH + ch] = Wd;
}

// B2: per-channel serial backward scan over chunk transforms -> anchors.
__global__ __launch_bounds__(256)
void bwd_boundary_kernel(const float* __restrict__ abuf,
                         const float* __restrict__ N,
                         float* __restrict__ anchors)  // [CC][NCH]
{
    const int ch = blockIdx.x * 256 + threadIdx.x;
    const int h  = ch & (Hh - 1);
    const int b  = ch >> 8;
    const size_t base = (size_t)b * Tt * Hh + h;
    float anc = abuf[base + (size_t)(Tt - 1) * Hh];    // h_{T-1} = a_{T-1}
    for (int c = CC - 1; c >= 0; --c) {
        anchors[c * NCH + ch] = anc;
        float U  = N[0 * CC * NCH + c * NCH + ch];
        float V  = N[1 * CC * NCH + c * NCH + ch];
        float Wd = N[2 * CC * NCH + c * NCH + ch];
        anc = (U * anc + V) * frcp(Wd);
    }
}

// B3: per-(chunk,channel) exact backward recurrence from the chunk anchor.
__global__ __launch_bounds__(256)
void bwd_apply_kernel(const float* __restrict__ probs,
                      const float* __restrict__ pbuf,
                      const float* __restrict__ abuf,
                      const float* __restrict__ anchors,
                      float* __restrict__ hout)
{
    const int tid = blockIdx.x * 256 + threadIdx.x;
    const int ch  = tid & (NCH - 1);
    const int c   = tid >> 13;
    const int h   = ch & (Hh - 1);
    const int b   = ch >> 8;
    const size_t base = (size_t)b * Tt * Hh + h;
    const float p1 = sigm(probs[Hh + h]);
    const float q1 = 1.0f - p1;

    float ht = anchors[c * NCH + ch];
    const int tlo = c * LL;
    int ttop;
    if (c == CC - 1) {
        hout[base + (size_t)(Tt - 1) * Hh] = abuf[base + (size_t)(Tt - 1) * Hh];
        ttop = Tt - 2;
    } else {
        ttop = (c + 1) * LL - 1;
    }
    for (int t = ttop; t >= tlo; --t) {
        const size_t idx = base + (size_t)t * Hh;
        if (t - 8 >= tlo)
            __builtin_prefetch(&abuf[idx - (size_t)8 * Hh], 0, 0);
        float a_t = abuf[idx];
        float pn  = pbuf[idx + Hh];
        ht = a_t * (p1 * ht * frcp(pn) + q1 * (1.0f - ht) * frcp(1.0f - pn));
        hout[idx] = ht;
    }
}

// ---------------------------------------------------------------------------
extern "C" void kernel_launch(void* const* d_in, const int* in_sizes, int n_in,
                              void* d_out, int out_size, void* d_ws, size_t ws_size,
                              hipStream_t stream)
{
    (void)in_sizes; (void)n_in; (void)out_size; (void)ws_size;

    const float* x      = (const float*)d_in[0];
    const float* W0     = (const float*)d_in[1];
    const float* W1     = (const float*)d_in[2];
    const float* probs0 = (const float*)d_in[3];
    const float* probs1 = (const float*)d_in[4];
    const float* g0     = (const float*)d_in[5];
    const float* b0     = (const float*)d_in[6];
    const float* g1     = (const float*)d_in[7];
    const float* b1     = (const float*)d_in[8];
    float* out = (float*)d_out;

    const size_t Nel = (size_t)Bb * Tt * Hh;   // 16,777,216 (64 MB)
    float* r    = (float*)d_ws;
    float* p    = r + Nel;
    float* a    = p + Nel;
    float* hm   = a + Nel;
    float* Mbuf = hm + Nel;                    // 4 * CC * NCH
    float* astart  = Mbuf + (size_t)4 * CC * NCH;
    float* Nbuf    = astart + (size_t)CC * NCH;  // 3 * CC * NCH
    float* anchors = Nbuf + (size_t)3 * CC * NCH;

    const int gemm_blocks  = MROWS / 128;                     // 512
    const size_t ldsBytes  = (size_t)Hh * Dd * sizeof(float); // 256 KB
    const int chunk_blocks = (CC * NCH) / 256;                // 2048
    const int chan_blocks  = NCH / 256;                       // 32

    // ---- Layer 1 ----
    gemm_ln_exp_kernel<<<gemm_blocks, 256, ldsBytes, stream>>>(x, W0, g0, b0, r);
    fwd_compose_kernel<<<chunk_blocks, 256, 0, stream>>>(r, probs0, Mbuf);
    fwd_boundary_kernel<<<chan_blocks, 256, 0, stream>>>(probs0, Mbuf, astart);
    fwd_apply_bwd_compose_kernel<<<chunk_blocks, 256, 0, stream>>>(r, probs0, astart, p, a, Nbuf);
    bwd_boundary_kernel<<<chan_blocks, 256, 0, stream>>>(a, Nbuf, anchors);
    bwd_apply_kernel<<<chunk_blocks, 256, 0, stream>>>(probs0, p, a, anchors, hm);

    // ---- Layer 2 ----
    gemm_ln_exp_kernel<<<gemm_blocks, 256, ldsBytes, stream>>>(hm, W1, g1, b1, r);
    fwd_compose_kernel<<<chunk_blocks, 256, 0, stream>>>(r, probs1, Mbuf);
    fwd_boundary_kernel<<<chan_blocks, 256, 0, stream>>>(probs1, Mbuf, astart);
    fwd_apply_bwd_compose_kernel<<<chunk_blocks, 256, 0, stream>>>(r, probs1, astart, p, a, Nbuf);
    bwd_boundary_kernel<<<chan_blocks, 256, 0, stream>>>(a, Nbuf, anchors);
    bwd_apply_kernel<<<chunk_blocks, 256, 0, stream>>>(probs1, p, a, anchors, out);
}